// RNN_28286654612083
// MI455X (gfx1250) — compile-verified
//
#include <hip/hip_runtime.h>
#include <math.h>

// Elman RNN (B=64, T=512, NIN=256, NH=1024, NOUT=256), all f32.
// Phase 1: xW = x @ W_ih^T + b_ih + b_hh            (WMMA f32 GEMM)
// Phase 2: persistent scan kernel, W_hh slice LDS-resident (async staged),
//          8 waves/WG with K-split + LDS reduction, per-step inter-WG sync
//          via monotonic atomic barrier (4 batch groups x 16 H-chunks)
// Phase 3: out = hs @ W_out^T + b_out               (WMMA f32 GEMM)
// Workspace requirement: 256 B (barriers) + 64*512*1024*4 = 128 MB (xW/hs in place).

typedef __attribute__((ext_vector_type(2))) float v2f;
typedef __attribute__((ext_vector_type(8))) float v8f;
typedef __attribute__((ext_vector_type(4))) int   v4i;

#define B_   64
#define T_   512
#define NIN  256
#define NH   1024
#define NOUT 256
#define BT   (B_ * T_)

#if defined(__has_builtin)
#if __has_builtin(__builtin_amdgcn_global_load_async_to_lds_b128)
#define ASYNC_LDS 1
#endif
#endif
#ifndef ASYNC_LDS
#define ASYNC_LDS 0
#endif

// 16-byte global -> LDS copy: async (ASYNCcnt-tracked, no VGPR round trip)
// when the gfx1250 builtin is available, else a plain staged store.
// Toolchain signature: (v4i* global_src, v4i* lds_dst, imm offset, imm cpol).
__device__ __forceinline__ void stage16(float* lds_dst, const float* gsrc) {
#if ASYNC_LDS
  __builtin_amdgcn_global_load_async_to_lds_b128((v4i*)gsrc, (v4i*)lds_dst, 0,
                                                 0);
#else
  *(float4*)lds_dst = *(const float4*)gsrc;
#endif
}

__device__ __forceinline__ void wait_stage() {
#if ASYNC_LDS
#if __has_builtin(__builtin_amdgcn_s_wait_asynccnt)
  __builtin_amdgcn_s_wait_asynccnt(0);
#else
  asm volatile("s_wait_asynccnt 0" ::: "memory");
#endif
#endif
}

__device__ __forceinline__ v8f wmma4(v2f a, v2f b, v8f c) {
  // V_WMMA_F32_16X16X4_F32: D = A(16x4,f32) * B(4x16,f32) + C(16x16,f32)
  return __builtin_amdgcn_wmma_f32_16x16x4_f32(false, a, false, b, (short)0, c,
                                               false, false);
}

__device__ __forceinline__ v8f v8zero() {
  v8f z = {0.f, 0.f, 0.f, 0.f, 0.f, 0.f, 0.f, 0.f};
  return z;
}

// ---------------------------------------------------------------------------
// Generic C[M,N] = A[M,K] @ W[N,K]^T + bias1 + bias2, f32 WMMA.
// Block tile: 16(M) x 64(N); 4 waves, each wave owns one 16x16 N-subtile.
// A tile staged in LDS with stride K+4 (bank-conflict-free for the frag reads).
// ---------------------------------------------------------------------------
template <int K>
__global__ void gemm_nt_wmma(const float* __restrict__ A,
                             const float* __restrict__ W,
                             const float* __restrict__ bias1,
                             const float* __restrict__ bias2,
                             float* __restrict__ C, int N) {
  constexpr int LDA = K + 4;  // (K+4) % 64 == 4 -> conflict-free b64 frag loads
  extern __shared__ float sA[];  // 16 x LDA floats

  const int tid  = threadIdx.x;   // 128 threads = 4 wave32
  const int wave = tid >> 5;
  const int lane = tid & 31;
  const int lr   = lane & 15;
  const int lhalf = (lane < 16) ? 0 : 2;

  const int m0 = blockIdx.x * 16;
  const int n0 = blockIdx.y * 64 + wave * 16;

  // Stage the 16 x K A-tile (async global->LDS, 16B per op, coalesced per row).
  {
    const int row = tid >> 3;            // 16 rows, 8 threads each
    const int per = K / 8;               // contiguous floats per thread
    const int c0  = (tid & 7) * per;
    const float* src = A + (size_t)(m0 + row) * K + c0;
    float* dst = sA + row * LDA + c0;
#pragma unroll
    for (int c = 0; c < per; c += 4) stage16(dst + c, src + c);
  }
  wait_stage();
  __syncthreads();

  const float* wrow = W + (size_t)(n0 + lr) * K;  // row of weight = B column

  v8f acc0 = v8zero(), acc1 = v8zero(), acc2 = v8zero(), acc3 = v8zero();
#pragma unroll 2
  for (int k = 0; k < K; k += 16) {
    v2f a0 = *(const v2f*)(sA + lr * LDA + k + 0 + lhalf);
    v2f a1 = *(const v2f*)(sA + lr * LDA + k + 4 + lhalf);
    v2f a2 = *(const v2f*)(sA + lr * LDA + k + 8 + lhalf);
    v2f a3 = *(const v2f*)(sA + lr * LDA + k + 12 + lhalf);
    v2f b0 = *(const v2f*)(wrow + k + 0 + lhalf);
    v2f b1 = *(const v2f*)(wrow + k + 4 + lhalf);
    v2f b2 = *(const v2f*)(wrow + k + 8 + lhalf);
    v2f b3 = *(const v2f*)(wrow + k + 12 + lhalf);
    acc0 = wmma4(a0, b0, acc0);
    acc1 = wmma4(a1, b1, acc1);
    acc2 = wmma4(a2, b2, acc2);
    acc3 = wmma4(a3, b3, acc3);
  }
  v8f acc = (acc0 + acc1) + (acc2 + acc3);

  const int n = n0 + lr;
  float bias = (bias1 ? bias1[n] : 0.f) + (bias2 ? bias2[n] : 0.f);
  const int mbase = m0 + ((lane < 16) ? 0 : 8);
#pragma unroll
  for (int v = 0; v < 8; ++v)
    C[(size_t)(mbase + v) * N + n] = acc[v] + bias;
}

// ---------------------------------------------------------------------------
// Persistent recurrence kernel. Grid = 64 WGs x 256 threads (8 wave32).
//   blockIdx.x = bg*16 + nc : bg in [0,4) batch group (16 batches, WMMA M=16),
//                             nc in [0,16) 64-wide H-chunk.
// W_hh rows [nc*64, nc*64+64) staged once in LDS (stride 1028 -> conflict-free),
// reused for all 512 steps. hs overwrites xW in place.
// K-split: waves 0-3 accumulate k in [0,512), waves 4-7 k in [512,1024) over
// the same four 16x16 N-subtiles; partials merged through a 4 KB LDS buffer.
// Per-step sync across the 16 WGs of a batch group: monotonic atomic counter.
// ---------------------------------------------------------------------------
#define LDW  1028  // 1024 + 4
#define KSPL 512

__device__ __forceinline__ void bg_barrier(unsigned* __restrict__ bar, int bg,
                                           unsigned step, int tid) {
  __syncthreads();
  __threadfence();  // release this WG's hs stores to agent scope
  if (tid == 0) {
    __hip_atomic_fetch_add(&bar[bg], 1u, __ATOMIC_RELEASE,
                           __HIP_MEMORY_SCOPE_AGENT);
    const unsigned target = 16u * step;
    while (__hip_atomic_load(&bar[bg], __ATOMIC_ACQUIRE,
                             __HIP_MEMORY_SCOPE_AGENT) < target) {
      __builtin_amdgcn_s_sleep(2);
    }
  }
  __syncthreads();
  __threadfence();  // acquire: see the other 15 WGs' hs stores
}

__global__ void rnn_scan(const float* __restrict__ Whh,  // [NH][NH]
                         float* __restrict__ hs,         // [B][T][NH] xW->hs
                         unsigned* __restrict__ bar) {
  extern __shared__ float sW[];        // 64 x LDW floats + 4x32x8 reduction buf
  float* sRed = sW + 64 * LDW;         // [4][32][8] floats

  const int tid  = threadIdx.x;  // 256 threads = 8 wave32
  const int wave = tid >> 5;
  const int lane = tid & 31;
  const int lr    = lane & 15;
  const int lhalf = (lane < 16) ? 0 : 2;
  const int nsub  = wave & 3;    // which 16-wide N subtile
  const int khalf = wave >> 2;   // which half of K

  const int bg = blockIdx.x >> 4;   // batch group: batches [bg*16, bg*16+16)
  const int nc = blockIdx.x & 15;   // H chunk:     cols    [nc*64, nc*64+64)
  const int m  = bg * 16;
  const int myn = nc * 64 + nsub * 16 + lr;  // output column of this lane

  // Stage W_hh chunk: 64 rows x 1024 cols, once (async). 4 threads per row.
  {
    const int row = tid >> 2;
    const int c0  = (tid & 3) * 256;
    const float* src = Whh + (size_t)(nc * 64 + row) * NH + c0;
    float* dst = sW + row * LDW + c0;
#pragma unroll 8
    for (int c = 0; c < 256; c += 4) stage16(dst + c, src + c);
  }
  wait_stage();
  __syncthreads();

  const float* wk = sW + (nsub * 16 + lr) * LDW + khalf * KSPL + lhalf;

  // t = 0: h0 = 0 -> h = tanh(xW[:,0,:]) in place (waves 0-3 only).
  if (wave < 4) {
    const int mb = m + ((lane < 16) ? 0 : 8);
#pragma unroll
    for (int v = 0; v < 8; ++v) {
      const size_t idx = ((size_t)(mb + v) * T_ + 0) * NH + myn;
      hs[idx] = tanhf(hs[idx]);
    }
  }
  bg_barrier(bar, bg, 1u, tid);

  for (int t = 1; t < T_; ++t) {
    // A = h_{t-1}[bg block] (16 x 1024, L2-resident hs), B = W_hh^T (LDS).
    const float* hprev =
        hs + ((size_t)(m + lr) * T_ + (t - 1)) * NH + khalf * KSPL + lhalf;

    v8f acc0 = v8zero(), acc1 = v8zero(), acc2 = v8zero(), acc3 = v8zero();
#pragma unroll 2
    for (int k = 0; k < KSPL; k += 16) {
      v2f a0 = *(const v2f*)(hprev + k + 0);
      v2f a1 = *(const v2f*)(hprev + k + 4);
      v2f a2 = *(const v2f*)(hprev + k + 8);
      v2f a3 = *(const v2f*)(hprev + k + 12);
      v2f b0 = *(const v2f*)(wk + k + 0);
      v2f b1 = *(const v2f*)(wk + k + 4);
      v2f b2 = *(const v2f*)(wk + k + 8);
      v2f b3 = *(const v2f*)(wk + k + 12);
      acc0 = wmma4(a0, b0, acc0);
      acc1 = wmma4(a1, b1, acc1);
      acc2 = wmma4(a2, b2, acc2);
      acc3 = wmma4(a3, b3, acc3);
    }
    v8f acc = (acc0 + acc1) + (acc2 + acc3);

    // Merge K-halves: waves 4-7 publish partials, waves 0-3 reduce + finish.
    if (wave >= 4) *(v8f*)(sRed + ((size_t)(wave - 4) * 32 + lane) * 8) = acc;
    __syncthreads();
    if (wave < 4) {
      acc = acc + *(const v8f*)(sRed + ((size_t)nsub * 32 + lane) * 8);
      const int mb = m + ((lane < 16) ? 0 : 8);
#pragma unroll
      for (int v = 0; v < 8; ++v) {
        const size_t idx = ((size_t)(mb + v) * T_ + t) * NH + myn;
        hs[idx] = tanhf(hs[idx] + acc[v]);  // xW + h@Whh^T, in place -> hs
      }
    }
    bg_barrier(bar, bg, (unsigned)(t + 1), tid);
  }
}

__global__ void init_bar_kernel(unsigned* bar) {
  if (threadIdx.x < 16) bar[threadIdx.x] = 0u;
}

// ---------------------------------------------------------------------------
extern "C" void kernel_launch(void* const* d_in, const int* in_sizes, int n_in,
                              void* d_out, int out_size, void* d_ws,
                              size_t ws_size, hipStream_t stream) {
  const float* x     = (const float*)d_in[0];  // [64,512,256]
  const float* W_ih  = (const float*)d_in[1];  // [1024,256]
  const float* W_hh  = (const float*)d_in[2];  // [1024,1024]
  const float* b_ih  = (const float*)d_in[3];  // [1024]
  const float* b_hh  = (const float*)d_in[4];  // [1024]
  const float* W_out = (const float*)d_in[5];  // [256,1024]
  const float* b_out = (const float*)d_in[6];  // [256]
  float* out = (float*)d_out;                  // [64,512,256]

  unsigned* bar = (unsigned*)d_ws;                    // 16 counters
  float* xw = (float*)((char*)d_ws + 256);            // [64][512][1024] = 128MB

  const int lds_g1  = (NIN + 4) * 16 * sizeof(float);           // 16,640 B
  const int lds_g3  = (NH + 4) * 16 * sizeof(float);            // 65,792 B
  const int lds_rnn = (64 * LDW + 4 * 32 * 8) * sizeof(float);  // 267,264 B

  (void)hipFuncSetAttribute((const void*)&gemm_nt_wmma<NH>,
                            hipFuncAttributeMaxDynamicSharedMemorySize, lds_g3);
  (void)hipFuncSetAttribute((const void*)&rnn_scan,
                            hipFuncAttributeMaxDynamicSharedMemorySize, lds_rnn);

  // Reset inter-WG barrier counters (deterministic per call).
  init_bar_kernel<<<1, 32, 0, stream>>>(bar);

  // Phase 1: xW = x @ W_ih^T + (b_ih + b_hh).  M=32768, N=1024, K=256.
  gemm_nt_wmma<NIN><<<dim3(BT / 16, NH / 64), 128, lds_g1, stream>>>(
      x, W_ih, b_ih, b_hh, xw, NH);

  // Phase 2: sequential scan, hs overwrites xW in place.
  rnn_scan<<<dim3(64), 256, lds_rnn, stream>>>(W_hh, xw, bar);

  // Phase 3: out = hs @ W_out^T + b_out.  M=32768, N=256, K=1024.
  gemm_nt_wmma<NH><<<dim3(BT / 16, NOUT / 64), 128, lds_g3, stream>>>(
      xw, W_out, b_out, nullptr, out, NOUT);
}